// SumLayer_47828755808817
// MI455X (gfx1250) — compile-verified
//
#include <hip/hip_runtime.h>

// ---------------------------------------------------------------------------
// SumLayer forward: weighted logsumexp over E=32 gathered children per node.
//   out[nids[n], b] = log(max(sum_e params[pids[n,e]] * exp(em[cids[n,e],b] - m), 1e-10)) + m
// CDNA5 path: TDM gather-mode async loads (tensor_load_to_lds) stage the 32
// scattered 1KB child rows per node into LDS, double-buffered, while waves
// run the exp/fma pipeline on the previous node.
// ---------------------------------------------------------------------------

#define N_NODES 32768
#define E       32
#define B       256
#define NPB     8          // nodes per block (pipeline depth along the loop)

typedef unsigned int u32x4 __attribute__((ext_vector_type(4)));
typedef int          i32x4 __attribute__((ext_vector_type(4)));
typedef int          i32x8 __attribute__((ext_vector_type(8)));

// Issue one gather-mode TDM load: 16 rows of element_mars (selected by 16-bit
// row indices) -> LDS at lds_byte. Each row = 256 f32 = 1KB, contiguous in LDS.
// D# layout per CDNA5 ISA (groups 0..3).
__device__ __forceinline__ void tdm_gather16(const float* __restrict__ em,
                                             unsigned lds_byte,
                                             const int* __restrict__ cid16)
{
  unsigned long long ga = (unsigned long long)em;

  u32x4 g0;
  g0[0] = 0x80000001u;                                   // count=1, gather_mode=1, 16b idx
  g0[1] = lds_byte;                                      // lds_addr (bytes)
  g0[2] = (unsigned)(ga & 0xFFFFFFFFull);                // global_addr[31:0]
  g0[3] = (unsigned)((ga >> 32) & 0x01FFFFFFull)         // global_addr[56:32]
        | 0x80000000u;                                   // type=2 ("image")

  i32x8 g1;
  g1[0] = 0x00020000;                                    // wg_mask=0, data_size=2 (4B)
  g1[1] = (int)(256u << 16);                             // tensor_dim0 = 256 (bits 79:48)
  g1[2] = 0;                                             // dim0 hi / tensor_dim1 lo
  g1[3] = (int)(1u | (256u << 16));                      // tensor_dim1=65536 hi, tile_dim0=256
  g1[4] = 16;                                            // tile_dim1 = #valid indices = 16
  g1[5] = 256;                                           // tensor_dim0_stride = 256 elems
  g1[6] = 0;
  g1[7] = 0;                                             // dim1_stride ignored in gather mode

  i32x4 g2, g3;                                          // row_index_0..15, packed 16-bit
#pragma unroll
  for (int j = 0; j < 4; ++j) {
    int a = cid16[2 * j];
    int b = cid16[2 * j + 1];
    g2[j] = (a & 0xFFFF) | (b << 16);
    int c = cid16[8 + 2 * j];
    int d = cid16[8 + 2 * j + 1];
    g3[j] = (c & 0xFFFF) | (d << 16);
  }

#if defined(__clang_major__) && (__clang_major__ >= 23)
  i32x8 gz = {0, 0, 0, 0, 0, 0, 0, 0};
  __builtin_amdgcn_tensor_load_to_lds(g0, g1, g2, g3, gz, 0);
#else
  __builtin_amdgcn_tensor_load_to_lds(g0, g1, g2, g3, 0);
#endif
}

__global__ __launch_bounds__(256) void sum_layer_tdm(
    const float* __restrict__ element_mars,   // [MAX_ELS, B]
    const float* __restrict__ params,         // [NUM_PARAMS]
    const int*   __restrict__ nids,           // [N]
    const int*   __restrict__ cids,           // [N, E]
    const int*   __restrict__ pids,           // [N, E]
    float*       __restrict__ out)            // [N, B]
{
  __shared__ float smem[2 * E * B];           // 2 x 32KB double buffer
  const int  t      = threadIdx.x;            // batch column
  const int  node0  = blockIdx.x * NPB;
  const bool issuer = (t < 32);               // wave 0 owns all TDM traffic

  const unsigned lds_base =
      (unsigned)(unsigned long long)&smem[0]; // flat addr low 32 = LDS byte offset

  // Prologue: prefetch node0's 32 child rows into buffer 0 (two 16-row gathers).
  if (issuer) {
    const int* c = cids + (long long)node0 * E;
    tdm_gather16(element_mars, lds_base,             c);
    tdm_gather16(element_mars, lds_base + 16 * 1024, c + 16);
  }

  for (int i = 0; i < NPB; ++i) {
    const int n   = node0 + i;
    const int buf = i & 1;

    if (issuer) __builtin_amdgcn_s_wait_tensorcnt(0);   // tile for node i landed
    __syncthreads();                                    // publish tile; prior buf^1 reads done

    // Prefetch next node's tile into the other buffer while we compute.
    if (issuer && (i + 1) < NPB) {
      const int* c  = cids + (long long)(n + 1) * E;
      unsigned   lb = lds_base + (unsigned)((buf ^ 1) * (E * B * 4));
      tdm_gather16(element_mars, lb,             c);
      tdm_gather16(element_mars, lb + 16 * 1024, c + 16);
    }

    // ---- weighted logsumexp over E=32 children, one batch column per thread
    const float* bp = &smem[buf * E * B];
    float x[E];
#pragma unroll
    for (int e = 0; e < E; ++e) x[e] = bp[e * B + t];   // bank-conflict-free ds loads

    float m = x[0];
#pragma unroll
    for (int e = 1; e < E; ++e) m = fmaxf(m, x[e]);

    const int* pp = pids + (long long)n * E;            // block-uniform -> s_loads
    float s = 0.0f;
#pragma unroll
    for (int e = 0; e < E; ++e) {
      float w = params[pp[e]];                          // block-uniform scalar gather
      s = fmaf(w, __expf(x[e] - m), s);                 // v_exp_f32 pipeline
    }

    float val = __logf(fmaxf(s, 1e-10f)) + m;
    out[(long long)nids[n] * B + t] = val;              // coalesced row store
  }
}

extern "C" void kernel_launch(void* const* d_in, const int* in_sizes, int n_in,
                              void* d_out, int out_size, void* d_ws, size_t ws_size,
                              hipStream_t stream)
{
  const float* node_mars    = (const float*)d_in[0];
  const float* element_mars = (const float*)d_in[1];
  const float* params       = (const float*)d_in[2];
  const int*   nids         = (const int*)d_in[3];
  const int*   cids         = (const int*)d_in[4];
  const int*   pids         = (const int*)d_in[5];
  float*       out          = (float*)d_out;

  // Seed output with node_mars (scatter semantics; nids covers all rows here,
  // but this keeps the kernel correct for any nids and is graph-capture safe).
  hipMemcpyAsync(out, node_mars, sizeof(float) * (size_t)N_NODES * B,
                 hipMemcpyDeviceToDevice, stream);

  sum_layer_tdm<<<N_NODES / NPB, 256, 0, stream>>>(element_mars, params, nids,
                                                   cids, pids, out);
}